// TWLayer_78219944395262
// MI455X (gfx1250) — compile-verified
//
#include <hip/hip_runtime.h>

// MI455X / gfx1250 fused tensor-wheel layer.
// Grid: one workgroup per batch element; 256 threads (8 wave32's).
// All intermediates LDS-resident; central GEMM on V_WMMA_F32_16X16X4_F32.

typedef __attribute__((ext_vector_type(2))) float v2f;
typedef __attribute__((ext_vector_type(8))) float v8f;

#define BATCH   512
#define NIN     10
#define NOUT    10
#define RNK     10
#define DIN     1024          // 2^NIN
#define DOUT    1024          // 2^NOUT
#define CORE_SZ (RNK*2*2*RNK) // 400 floats per core
#define SWEEP   (RNK*DIN)     // 10240 floats of sweep state per (b,l)
#define NTHREADS 256

__global__ __launch_bounds__(NTHREADS)
void tw_fused_kernel(const float* __restrict__ x,        // (B, 1024)
                     const float* __restrict__ icores,   // (10,10,2,2,10)
                     const float* __restrict__ ocores,   // (10,10,2,2,10)
                     const float* __restrict__ W,        // (1024,1024) central core
                     const float* __restrict__ bias,     // (1,1024)
                     float* __restrict__ out)            // (B,1024)
{
    __shared__ float xs[DIN];                 //  4 KB
    __shared__ float cin[NIN * CORE_SZ];      // 16 KB
    __shared__ float cot[NOUT * CORE_SZ];     // 16 KB
    __shared__ float bufA[SWEEP];             // 40 KB  (ping)
    __shared__ float bufB[SWEEP];             // 40 KB  (pong)
    __shared__ float oacc[DOUT];              //  4 KB  (output accumulator)

    const int b   = blockIdx.x;
    const int tid = threadIdx.x;

    // ---- stage inputs into LDS; init output accumulator with bias ----
    for (int i = tid; i < DIN; i += NTHREADS)            xs[i]  = x[(size_t)b * DIN + i];
    for (int i = tid; i < NIN * CORE_SZ; i += NTHREADS)  cin[i] = icores[i];
    for (int i = tid; i < NOUT * CORE_SZ; i += NTHREADS) cot[i] = ocores[i];
    for (int i = tid; i < DOUT; i += NTHREADS)           oacc[i] = bias[i];
    __syncthreads();

    const int lane = tid & 31;
    const int wave = tid >> 5;
    const int m16  = lane & 15;            // WMMA row (M) / B column offset
    const int kh   = (lane >> 4) << 1;     // K half-select: 0 or 2
    // Clamp A row: D rows >= RNK are never stored, so rows 10..15 may carry
    // duplicates of row 0 -- this keeps the LDS loads UNCONDITIONAL (no exec
    // masking around ds_load) while staying in-bounds.
    const int mr   = (m16 < RNK) ? m16 : 0;

    for (int l = 0; l < RNK; ++l) {
        // ================= input sweep =================
        // step 0: S0[c,s,q] = sum_p xs[p*512+q] * core0[l,p,c,s]
        // state layout: buf[(c*RNK + r)*Q + q]
        for (int o = tid; o < 2 * RNK * (DIN / 2); o += NTHREADS) {
            int q = o & 511;
            int t = o >> 9;
            int s = t % RNK;
            int c = t / RNK;                         // 0..1
            float a = xs[q]       * cin[((l * 2 + 0) * 2 + c) * RNK + s]
                    + xs[512 + q] * cin[((l * 2 + 1) * 2 + c) * RNK + s];
            bufA[(c * RNK + s) * 512 + q] = a;
        }
        __syncthreads();

        float* src = bufA;
        float* dst = bufB;
        for (int i = 1; i < NIN; ++i) {
            const int shift = NIN - i - 1;           // log2(Q/2)
            const int Qh = 1 << shift;
            const int Q  = Qh << 1;
            const float* ci = &cin[i * CORE_SZ];
            for (int o = tid; o < SWEEP; o += NTHREADS) {
                int qr = o & (Qh - 1);
                int t  = o >> shift;
                int s  = t % RNK;
                int u  = t / RNK;
                int d  = u & 1;
                int c  = u >> 1;
                const float* inrow = &src[c * RNK * Q + qr];
                float a = 0.f;
                #pragma unroll
                for (int r = 0; r < RNK; ++r) {
                    a += inrow[r * Q]      * ci[((r * 2 + 0) * 2 + d) * RNK + s];
                    a += inrow[r * Q + Qh] * ci[((r * 2 + 1) * 2 + d) * RNK + s];
                }
                dst[((c * 2 + d) * RNK + s) * Qh + qr] = a;
            }
            __syncthreads();
            float* tmp = src; src = dst; dst = tmp;
        }
        // final S in `src`: S[c][r] at src[c*RNK + r]   (9 steps -> src == bufB)

        // ================= central GEMM (fp32 WMMA) =================
        // U[r][d] = sum_c S[c][r] * W[c][d];  M=r (10 of 16), N=d (1024), K=c (1024)
        // A 16x4 f32: lanes 0-15 hold K+0/K+1, lanes 16-31 hold K+2/K+3, M=lane%16
        // B 4x16 f32: same K halves, N=lane%16
        // Loop order: outer K, inner 8 N-tiles -> A loaded once per K-step,
        // 8 independent WMMAs back-to-back for latency hiding.
        float* Ubuf = dst;                           // the non-S buffer
        const int ncol0 = wave * 128 + m16;          // this wave's first N column
        const float* ap = src + kh * RNK + mr;       // per-lane A base (K half)

        v8f acc[8];
        #pragma unroll
        for (int t8 = 0; t8 < 8; ++t8)
            acc[t8] = (v8f){0.f, 0.f, 0.f, 0.f, 0.f, 0.f, 0.f, 0.f};

        for (int k = 0; k < DIN; k += 4) {
            v2f a;
            a.x = ap[(k    ) * RNK];                 // unconditional ds_load_b32
            a.y = ap[(k + 1) * RNK];
            const float* wrow = &W[(size_t)(k + kh) * DOUT + ncol0];
            __builtin_prefetch(wrow + 4 * DOUT, 0, 3);   // global_prefetch_b8
            #pragma unroll
            for (int t8 = 0; t8 < 8; ++t8) {
                v2f bb;
                bb.x = wrow[t8 * 16];
                bb.y = wrow[t8 * 16 + DOUT];
                acc[t8] = __builtin_amdgcn_wmma_f32_16x16x4_f32(
                              false, a, false, bb, (short)0, acc[t8], false, false);
            }
        }
        // D 16x16 f32: VGPR v holds M = v + (lane>=16 ? 8 : 0), N = column
        {
            const int mbase = (lane >> 4) * 8;
            #pragma unroll
            for (int t8 = 0; t8 < 8; ++t8) {
                const int n = ncol0 + t8 * 16;
                #pragma unroll
                for (int v = 0; v < 8; ++v) {
                    const int M = mbase + v;
                    if (M < RNK) Ubuf[M * DOUT + n] = acc[t8][v];  // V0: [x=0][r][D]
                }
            }
        }
        __syncthreads();

        // ================= output sweep =================
        float* vsrc = Ubuf;
        float* vdst = src;                           // S no longer needed
        for (int j = 0; j < NOUT; ++j) {
            const int shift = NOUT - j - 1;
            const int Dh = 1 << shift;
            const int D  = Dh << 1;
            const float* cj = &cot[j * CORE_SZ];
            for (int o = tid; o < SWEEP; o += NTHREADS) {
                int dq = o & (Dh - 1);
                int t  = o >> shift;
                int s  = t % RNK;
                int u  = t / RNK;
                int p  = u & 1;
                int xx = u >> 1;
                const float* inrow = &vsrc[xx * RNK * D + dq];
                float a = 0.f;
                #pragma unroll
                for (int r = 0; r < RNK; ++r) {
                    a += inrow[r * D]      * cj[((r * 2 + p) * 2 + 0) * RNK + s];
                    a += inrow[r * D + Dh] * cj[((r * 2 + p) * 2 + 1) * RNK + s];
                }
                vdst[((xx * 2 + p) * RNK + s) * Dh + dq] = a;
            }
            __syncthreads();
            float* tmp = vsrc; vsrc = vdst; vdst = tmp;
        }
        // final V in vsrc: V[p][r] at vsrc[p*RNK + r]; close the ring with r = l
        for (int p = tid; p < DOUT; p += NTHREADS)
            oacc[p] += vsrc[p * RNK + l];
        __syncthreads();
    }

    for (int i = tid; i < DOUT; i += NTHREADS)
        out[(size_t)b * DOUT + i] = oacc[i];
}

extern "C" void kernel_launch(void* const* d_in, const int* in_sizes, int n_in,
                              void* d_out, int out_size, void* d_ws, size_t ws_size,
                              hipStream_t stream)
{
    (void)in_sizes; (void)n_in; (void)d_ws; (void)ws_size; (void)out_size;
    const float* x      = (const float*)d_in[0];
    const float* icores = (const float*)d_in[1];
    const float* ocores = (const float*)d_in[2];
    const float* W      = (const float*)d_in[3];
    const float* bias   = (const float*)d_in[4];
    float* out = (float*)d_out;

    tw_fused_kernel<<<dim3(BATCH), dim3(NTHREADS), 0, stream>>>(
        x, icores, ocores, W, bias, out);
}